// NeuroCoreLayer_27144193311173
// MI455X (gfx1250) — compile-verified
//
#include <hip/hip_runtime.h>
#include <hip/hip_bf16.h>

#define D 128

typedef float v2f __attribute__((ext_vector_type(2)));
typedef float v8f __attribute__((ext_vector_type(8)));

enum { P_PLAIN = 0, P_CUP = 1, P_LUP = 2 };
enum { E_STORE = 0, E_SCATTER = 1 };

__device__ __forceinline__ v8f wmma_f32_4(v2f a, v2f b, v8f c) {
  // V_WMMA_F32_16X16X4_F32 : D = A(16x4,f32) x B(4x16,f32) + C(16x16,f32)
  return __builtin_amdgcn_wmma_f32_16x16x4_f32(false, a, false, b, (short)0, c,
                                               false, false);
}

__device__ __forceinline__ v8f bcast8(float v) {
  v8f a;
#pragma unroll
  for (int r = 0; r < 8; ++r) a[r] = v;
  return a;
}

// MT row-tiles (16 rows each) x one 16-col tile per wave.
// X in LDS row-major [16*MT][ldx]; W global [K][128].
// Each B fragment load is reused by MT wmma ops (amortizes weight traffic).
template <int K, int MT>
__device__ __forceinline__ void gemm_tiles(const float* Xs, int ldx,
                                           const float* __restrict__ W,
                                           int ncol0, int lane, v8f acc[MT]) {
  const int half = lane >> 4, lq = lane & 15;
  const float* wcol = W + 2 * half * D + ncol0 + lq;  // b.x at W[k+2h][n]
#pragma unroll 4
  for (int k = 0; k < K; k += 4) {
    v2f b; b.x = wcol[k * D]; b.y = wcol[k * D + D];
#pragma unroll
    for (int m = 0; m < MT; ++m) {
      const float* xrow = Xs + (16 * m + lq) * ldx + 2 * half;
      v2f a; a.x = xrow[k]; a.y = xrow[k + 1];
      acc[m] = wmma_f32_4(a, b, acc[m]);
    }
  }
}

// Fused 3-layer MLP: out = (relu(relu(X@W1+b0)@Wh0+b1))@Wh1+b2
// Block = 256 threads = 8 wave32; wave w owns columns [16w,16w+16) of
// 16*MT rows.
template <int K1, int MT, int PROLOG, int EPI>
__global__ void __launch_bounds__(256)
mlp3_kernel(const float* __restrict__ xbase,    // P_PLAIN: [rows][128]
            const float* __restrict__ embp,     // P_CUP: emb+nL*D ; P_LUP: emb
            const float* __restrict__ gatherA,  // P_CUP: L ; P_LUP: cl
            const float* __restrict__ gatherB,  // P_LUP: L (flip)
            const int* __restrict__ edge_lit,
            const float* __restrict__ W1,       // [K1][128]
            const float* __restrict__ Wh,       // [2][128][128]
            const float* __restrict__ bias,     // [3][128]
            float* __restrict__ out,            // E_STORE target
            float* __restrict__ clbuf,          // E_SCATTER target [nL][128]
            int rows, int nV) {
  constexpr int RB = 16 * MT;   // rows per block
  constexpr int LDX = K1 + 4;   // padded strides -> conflict-free LDS
  constexpr int LDH = D + 4;
  __shared__ float Xs[RB * LDX];
  __shared__ float Hs[RB * LDH];
  __shared__ int el[3 * RB];

  const int tid = threadIdx.x;
  const int lane = tid & 31;
  const int wid = tid >> 5;
  const int row0 = blockIdx.x * RB;
  const int ncol0 = wid * 16;
  const int half = lane >> 4, lq = lane & 15;

  // ---------------- prologue: stage the RB x K1 activation tile -----------
  if (PROLOG == P_PLAIN) {
    for (int i = tid; i < RB * 32; i += 256) {
      int r = i >> 5, c = i & 31;
      float4 v = {0.f, 0.f, 0.f, 0.f};
      if (row0 + r < rows)
        v = ((const float4*)(xbase + (size_t)(row0 + r) * D))[c];
      *(float4*)(Xs + r * LDX + c * 4) = v;
    }
  } else if (PROLOG == P_CUP) {  // [ emb_clause | sum of 3 gathered L rows ]
    for (int i = tid; i < RB * 64; i += 256) {
      int r = i >> 6, c = i & 63;
      float4 v = {0.f, 0.f, 0.f, 0.f};
      int cc = row0 + r;
      if (cc < rows) {
        if (c < 32) {
          v = ((const float4*)(embp + (size_t)cc * D))[c];
        } else {
          int g = c - 32;
          int e0 = edge_lit[3 * cc + 0];
          int e1 = edge_lit[3 * cc + 1];
          int e2 = edge_lit[3 * cc + 2];
          float4 p = ((const float4*)(gatherA + (size_t)e0 * D))[g];
          float4 q = ((const float4*)(gatherA + (size_t)e1 * D))[g];
          float4 s = ((const float4*)(gatherA + (size_t)e2 * D))[g];
          v.x = p.x + q.x + s.x; v.y = p.y + q.y + s.y;
          v.z = p.z + q.z + s.z; v.w = p.w + q.w + s.w;
        }
      }
      *(float4*)(Xs + r * LDX + c * 4) = v;
    }
  } else {  // P_LUP: [ emb_lit | cl_msg | L[flip] ]
    for (int i = tid; i < RB * 96; i += 256) {
      int r = i / 96, c = i % 96;
      float4 v = {0.f, 0.f, 0.f, 0.f};
      int l = row0 + r;
      if (l < rows) {
        if (c < 32)
          v = ((const float4*)(embp + (size_t)l * D))[c];
        else if (c < 64)
          v = ((const float4*)(gatherA + (size_t)l * D))[c - 32];
        else {
          int lf = (l < nV) ? (l + nV) : (l - nV);
          v = ((const float4*)(gatherB + (size_t)lf * D))[c - 64];
        }
      }
      *(float4*)(Xs + r * LDX + c * 4) = v;
    }
  }
  if (EPI == E_SCATTER) {
    for (int i = tid; i < 3 * RB; i += 256) {
      int cc = row0 + i / 3;
      el[i] = (cc < rows) ? edge_lit[3 * row0 + i] : -1;
    }
  }
  __syncthreads();

  // ---------------- layer 1 : X @ W1 + b0, relu -> Hs ---------------------
  v8f acc[MT];
#pragma unroll
  for (int m = 0; m < MT; ++m) acc[m] = bcast8(bias[ncol0 + lq]);
  gemm_tiles<K1, MT>(Xs, LDX, W1, ncol0, lane, acc);
#pragma unroll
  for (int m = 0; m < MT; ++m)
#pragma unroll
    for (int r = 0; r < 8; ++r)
      Hs[(16 * m + r + 8 * half) * LDH + ncol0 + lq] = fmaxf(acc[m][r], 0.f);
  __syncthreads();  // all waves done reading Xs + writing Hs

  // ---------------- layer 2 : H @ Wh0 + b1, relu -> Xs (reused) -----------
#pragma unroll
  for (int m = 0; m < MT; ++m) acc[m] = bcast8(bias[D + ncol0 + lq]);
  gemm_tiles<D, MT>(Hs, LDH, Wh, ncol0, lane, acc);
#pragma unroll
  for (int m = 0; m < MT; ++m)
#pragma unroll
    for (int r = 0; r < 8; ++r)
      Xs[(16 * m + r + 8 * half) * LDH + ncol0 + lq] = fmaxf(acc[m][r], 0.f);
  __syncthreads();

  // ---------------- layer 3 : H2 @ Wh1 + b2 -------------------------------
#pragma unroll
  for (int m = 0; m < MT; ++m) acc[m] = bcast8(bias[2 * D + ncol0 + lq]);
  gemm_tiles<D, MT>(Xs, LDH, Wh + D * D, ncol0, lane, acc);

  // ---------------- epilogue ----------------------------------------------
  if (EPI == E_STORE) {
#pragma unroll
    for (int m = 0; m < MT; ++m)
#pragma unroll
      for (int r = 0; r < 8; ++r) {
        int rr = 16 * m + r + 8 * half;
        if (row0 + rr < rows)
          out[(size_t)(row0 + rr) * D + ncol0 + lq] = acc[m][r];
      }
  } else {  // scatter: cl[edge_lit[3c+j]] += Cm[c]   (j = 0..2)
#pragma unroll
    for (int m = 0; m < MT; ++m)
#pragma unroll
      for (int r = 0; r < 8; ++r) {
        int rr = 16 * m + r + 8 * half;
        float v = acc[m][r];
#pragma unroll
        for (int j = 0; j < 3; ++j) {
          int t = el[3 * rr + j];
          if (t >= 0)
            unsafeAtomicAdd(clbuf + (size_t)t * D + ncol0 + lq, v);
        }
      }
  }
}

__global__ void zero_f4(float4* p, long n4) {
  long i = (long)blockIdx.x * blockDim.x + threadIdx.x;
  long stride = (long)gridDim.x * blockDim.x;
  float4 z = {0.f, 0.f, 0.f, 0.f};
  for (; i < n4; i += stride) p[i] = z;
}

extern "C" void kernel_launch(void* const* d_in, const int* in_sizes, int n_in,
                              void* d_out, int out_size, void* d_ws,
                              size_t ws_size, hipStream_t stream) {
  const float* emb     = (const float*)d_in[0];
  const int*   edgeLit = (const int*)d_in[2];
  const float* LmsgW   = (const float*)d_in[6];
  const float* Lmsgb   = (const float*)d_in[7];
  const float* CmsgW   = (const float*)d_in[8];
  const float* Cmsgb   = (const float*)d_in[9];
  const float* CupW1   = (const float*)d_in[10];
  const float* CupWh   = (const float*)d_in[11];
  const float* Cupb    = (const float*)d_in[12];
  const float* LupW1   = (const float*)d_in[13];
  const float* LupWh   = (const float*)d_in[14];
  const float* Lupb    = (const float*)d_in[15];

  const int N  = in_sizes[0] / D;
  const int E  = in_sizes[2];
  const int C  = E / 3;
  const int nV = (N - C) / 2;
  const int nL = 2 * nV;

  float* out   = (float*)d_out;
  float* Lbuf  = (float*)d_ws;                   // [nL][128] literal messages
  float* clbuf = Lbuf + (size_t)nL * D;          // [nL][128] scattered accum

  constexpr int MT = 4;
  constexpr int RB = 16 * MT;
  dim3 blk(256);

  // 1) L = mlp3(emb[:nL], Lmsg)  -> ws.L
  mlp3_kernel<D, MT, P_PLAIN, E_STORE><<<(nL + RB - 1) / RB, blk, 0, stream>>>(
      emb, nullptr, nullptr, nullptr, edgeLit,
      LmsgW, LmsgW + D * D, Lmsgb, Lbuf, nullptr, nL, nV);

  // zero the cl accumulator (harness poisons ws)
  {
    long n4 = (long)nL * D / 4;
    int g = (int)((n4 + 255) / 256);
    if (g > 32768) g = 32768;
    zero_f4<<<g, blk, 0, stream>>>((float4*)clbuf, n4);
  }

  // 2) c_new = mlp3([emb_clause | gather3(L)], Cup) -> out[nL:]
  mlp3_kernel<2 * D, MT, P_CUP, E_STORE><<<(C + RB - 1) / RB, blk, 0, stream>>>(
      nullptr, emb + (size_t)nL * D, Lbuf, nullptr, edgeLit,
      CupW1, CupWh, Cupb, out + (size_t)nL * D, nullptr, C, nV);

  // 3) Cm = mlp3(c_new, Cmsg), scattered into cl on the fly
  mlp3_kernel<D, MT, P_PLAIN, E_SCATTER><<<(C + RB - 1) / RB, blk, 0, stream>>>(
      out + (size_t)nL * D, nullptr, nullptr, nullptr, edgeLit,
      CmsgW, CmsgW + D * D, Cmsgb, nullptr, clbuf, C, nV);

  // 4) l_new = mlp3([emb_lit | cl | L(flip)], Lup) -> out[:nL]
  mlp3_kernel<3 * D, MT, P_LUP, E_STORE><<<(nL + RB - 1) / RB, blk, 0, stream>>>(
      nullptr, emb, clbuf, Lbuf, edgeLit,
      LupW1, LupWh, Lupb, out, nullptr, nL, nV);
}